// DeformableSpatialAttention_1314259993168
// MI455X (gfx1250) — compile-verified
//
#include <hip/hip_runtime.h>
#include <math.h>

// Problem constants (from reference)
#define BATCH 16
#define CHANS 256
#define IMH 128
#define IMW 128
#define HWSZ (IMH * IMW)
#define KWIN 7
#define EPSV 1e-5f

typedef float v2f __attribute__((ext_vector_type(2)));
typedef float v4f __attribute__((ext_vector_type(4)));
typedef float v8f __attribute__((ext_vector_type(8)));

// ---------------------------------------------------------------------------
// Kernel 1: channel mean + max over 256 channels (bandwidth-dominant stage).
// 128-bit non-temporal streaming loads; each thread owns 4 consecutive pixels.
// ---------------------------------------------------------------------------
__global__ void k_chan_reduce(const float* __restrict__ x, float* __restrict__ sf) {
    const int q   = HWSZ / 4;                               // float4 slots per plane
    int tid = blockIdx.x * blockDim.x + threadIdx.x;        // BATCH * q = 65536
    int b   = tid / q;
    int p4  = tid - b * q;

    const v4f* xp = reinterpret_cast<const v4f*>(x + (size_t)b * CHANS * HWSZ) + p4;

    v4f s = (v4f)0.0f;
    v4f m = (v4f)(-3.402823466e38f);
    #pragma unroll 8
    for (int c = 0; c < CHANS; ++c) {
        v4f v = __builtin_nontemporal_load(xp + (size_t)c * q);
        s += v;
        m.x = fmaxf(m.x, v.x); m.y = fmaxf(m.y, v.y);
        m.z = fmaxf(m.z, v.z); m.w = fmaxf(m.w, v.w);
    }
    v4f avg = s * (1.0f / (float)CHANS);
    reinterpret_cast<v4f*>(sf + ((size_t)b * 2 + 0) * HWSZ)[p4] = avg;
    reinterpret_cast<v4f*>(sf + ((size_t)b * 2 + 1) * HWSZ)[p4] = m;
}

// ---------------------------------------------------------------------------
// Kernel 2: fold the mean-over-49 groups into conv2 weights (linearity).
// w_eff[2][16][3][3] = mean_k w2[d*49+k][:][:][:],  b_eff[d] = mean_k b2[d*49+k]
// ---------------------------------------------------------------------------
__global__ void k_fold_w2(const float* __restrict__ w2, const float* __restrict__ b2,
                          float* __restrict__ w_eff, float* __restrict__ b_eff) {
    int t = threadIdx.x;
    if (t < 288) {
        int d = t / 144, r = t - d * 144;
        float s = 0.0f;
        for (int k = 0; k < 49; ++k) s += w2[(size_t)(d * 49 + k) * 144 + r];
        w_eff[t] = s * (1.0f / 49.0f);
    } else if (t < 290) {
        int d = t - 288;
        float s = 0.0f;
        for (int k = 0; k < 49; ++k) s += b2[d * 49 + k];
        b_eff[d] = s * (1.0f / 49.0f);
    }
}

// ---------------------------------------------------------------------------
// Kernel 3: conv3x3 (2 -> 16) + folded BN + ReLU via V_WMMA_F32_16X16X4_F32.
// One wave computes a 16 (out-channel) x 16 (pixel) tile. K = 18 padded to 20
// -> 5 WMMA issues. fp32 WMMA keeps reference precision.
//
// A (16x4 fp32) layout: lane half h = lane>>4, row m = lane&15,
//   a[0] = A[m][4*chunk + 2h], a[1] = A[m][4*chunk + 2h + 1]
// B (4x16 fp32) layout: col n = lane&15, b[v] = B[4*chunk + v + 2h][n]
// C/D (16x16 fp32): c[v] = D[v + 8h][n]
//
// All tap-derived indices (ci, dh, dw, valid) are computed separately for the
// half=0 / half=1 k-values (unroll-time constants -> /9, /3 fold away), then
// selected with one cndmask. Loads use clamped in-bounds addresses + select-0
// so no exec-mask branching occurs between WMMA issues.
// ---------------------------------------------------------------------------
__global__ void k_conv1_wmma(const float* __restrict__ sf, const float* __restrict__ w1,
                             const float* __restrict__ bn_g, const float* __restrict__ bn_b,
                             const float* __restrict__ bn_m, const float* __restrict__ bn_v,
                             float* __restrict__ h1) {
    const int tilesPerRow = IMW / 16;                       // 8
    int wave = (blockIdx.x * blockDim.x + threadIdx.x) >> 5;
    int lane = threadIdx.x & 31;

    int b   = wave / (IMH * tilesPerRow);
    int rem = wave - b * (IMH * tilesPerRow);
    int hh  = rem / tilesPerRow;
    int w0  = (rem - hh * tilesPerRow) * 16;

    int  half  = lane >> 4;
    bool upper = (half != 0);
    int  mn    = lane & 15;                                 // A row m and B/C column n
    int  px    = w0 + mn;                                   // this lane's pixel column

    const float* sfb = sf + ((size_t)b * 2) * HWSZ;

    // BN scale folded into A rows (same m for both lane halves)
    float scale_m = bn_g[mn] * rsqrtf(bn_v[mn] + EPSV);

    v8f c = (v8f)0.0f;
    #pragma unroll
    for (int chunk = 0; chunk < 5; ++chunk) {
        // ---- A: weights, rows m, K = 4*chunk + 2*half + {0,1} -------------
        v2f a;
        {
            int ka  = chunk * 4 + 2 * half;
            int i0  = (ka     < 18) ? ka     : 0;           // clamped (always valid)
            int i1  = (ka + 1 < 18) ? ka + 1 : 0;
            float w0v = w1[mn * 18 + i0] * scale_m;
            float w1v = w1[mn * 18 + i1] * scale_m;
            a.x = (ka     < 18) ? w0v : 0.0f;
            a.y = (ka + 1 < 18) ? w1v : 0.0f;
        }

        // ---- B: im2col, K = 4*chunk + comp + 2*half, column = px ----------
        v2f bm;
        #pragma unroll
        for (int comp = 0; comp < 2; ++comp) {
            const int klo = chunk * 4 + comp;               // half == 0 k-value
            const int khi = klo + 2;                        // half == 1 k-value
            // constant-folding derived indices for both halves
            const int  ci_lo = (klo < 18) ? (klo / 9) : 0;
            const int  dh_lo = (klo < 18) ? ((klo % 9) / 3 - 1) : 0;
            const int  dw_lo = (klo < 18) ? ((klo % 9) % 3 - 1) : 0;
            const bool va_lo = (klo < 18);
            const int  ci_hi = (khi < 18) ? (khi / 9) : 0;
            const int  dh_hi = (khi < 18) ? ((khi % 9) / 3 - 1) : 0;
            const int  dw_hi = (khi < 18) ? ((khi % 9) % 3 - 1) : 0;
            const bool va_hi = (khi < 18);

            int  ci    = upper ? ci_hi : ci_lo;
            int  dh    = upper ? dh_hi : dh_lo;
            int  dw    = upper ? dw_hi : dw_lo;
            bool kval  = upper ? va_hi : va_lo;

            int hy = hh + dh;
            int wx = px + dw;
            bool inb = kval & ((unsigned)hy < (unsigned)IMH)
                            & ((unsigned)wx < (unsigned)IMW);
            int hyc = min(max(hy, 0), IMH - 1);
            int wxc = min(max(wx, 0), IMW - 1);
            float v = sfb[(size_t)ci * HWSZ + (size_t)hyc * IMW + wxc]; // always in-bounds
            float r = inb ? v : 0.0f;
            if (comp == 0) bm.x = r; else bm.y = r;
        }

        c = __builtin_amdgcn_wmma_f32_16x16x4_f32(false, a, false, bm,
                                                  (short)0, c, false, false);
    }

    // bias (folded BN shift) + ReLU + store: lane holds D[v + 8*half][n]
    #pragma unroll
    for (int v = 0; v < 8; ++v) {
        int M = v + 8 * half;
        float sM = bn_g[M] * rsqrtf(bn_v[M] + EPSV);
        float bM = bn_b[M] - bn_m[M] * sM;
        float val = fmaxf(c[v] + bM, 0.0f);
        h1[((size_t)(b * 16 + M)) * HWSZ + (size_t)hh * IMW + px] = val;
    }
}

// ---------------------------------------------------------------------------
// Kernel 4: folded conv3x3 (16 -> 2) + tanh*0.5 -> deform grid -> bilinear
// sample of sf (zero padding, align_corners=False). One thread per pixel.
// ---------------------------------------------------------------------------
__global__ void k_offset_sample(const float* __restrict__ h1,
                                const float* __restrict__ w_eff,
                                const float* __restrict__ b_eff,
                                const float* __restrict__ sf,
                                float* __restrict__ sampled) {
    __shared__ float swe[290];
    for (int t = threadIdx.x; t < 290; t += blockDim.x)
        swe[t] = (t < 288) ? w_eff[t] : b_eff[t - 288];
    __syncthreads();

    int tid = blockIdx.x * blockDim.x + threadIdx.x;        // BATCH*HWSZ
    int b  = tid / HWSZ;
    int p  = tid - b * HWSZ;
    int hh = p / IMW;
    int ww = p - hh * IMW;

    float acc0 = swe[288];
    float acc1 = swe[289];
    #pragma unroll
    for (int tap = 0; tap < 9; ++tap) {
        int y = hh + tap / 3 - 1;
        int x = ww + (tap % 3) - 1;
        bool ok = ((unsigned)y < (unsigned)IMH) && ((unsigned)x < (unsigned)IMW);
        int yc = min(max(y, 0), IMH - 1);
        int xc = min(max(x, 0), IMW - 1);
        #pragma unroll 4
        for (int ci = 0; ci < 16; ++ci) {
            float raw = h1[((size_t)(b * 16 + ci)) * HWSZ + (size_t)yc * IMW + xc];
            float v   = ok ? raw : 0.0f;
            acc0 = fmaf(v, swe[      ci * 9 + tap], acc0);
            acc1 = fmaf(v, swe[144 + ci * 9 + tap], acc1);
        }
    }

    float offx = tanhf(acc0) * 0.5f;
    float offy = tanhf(acc1) * 0.5f;

    float basex = -1.0f + 2.0f * (float)ww / (float)(IMW - 1);
    float basey = -1.0f + 2.0f * (float)hh / (float)(IMH - 1);
    float gx = fminf(fmaxf(basex + offx, -1.0f), 1.0f);
    float gy = fminf(fmaxf(basey + offy, -1.0f), 1.0f);

    float ix = ((gx + 1.0f) * (float)IMW - 1.0f) * 0.5f;
    float iy = ((gy + 1.0f) * (float)IMH - 1.0f) * 0.5f;
    float x0f = floorf(ix), y0f = floorf(iy);
    int x0 = (int)x0f, y0 = (int)y0f;
    int x1 = x0 + 1,  y1 = y0 + 1;
    float wx1 = ix - x0f, wx0 = 1.0f - wx1;
    float wy1 = iy - y0f, wy0 = 1.0f - wy1;

    int x0c = min(max(x0, 0), IMW - 1), x1c = min(max(x1, 0), IMW - 1);
    int y0c = min(max(y0, 0), IMH - 1), y1c = min(max(y1, 0), IMH - 1);
    bool vx0 = (unsigned)x0 < (unsigned)IMW, vx1 = (unsigned)x1 < (unsigned)IMW;
    bool vy0 = (unsigned)y0 < (unsigned)IMH, vy1 = (unsigned)y1 < (unsigned)IMH;

    #pragma unroll
    for (int c = 0; c < 2; ++c) {
        const float* fp = sf + ((size_t)(b * 2 + c)) * HWSZ;
        float v00 = (vx0 && vy0) ? fp[(size_t)y0c * IMW + x0c] : 0.0f;
        float v10 = (vx1 && vy0) ? fp[(size_t)y0c * IMW + x1c] : 0.0f;
        float v01 = (vx0 && vy1) ? fp[(size_t)y1c * IMW + x0c] : 0.0f;
        float v11 = (vx1 && vy1) ? fp[(size_t)y1c * IMW + x1c] : 0.0f;
        float out = v00 * (wx0 * wy0) + v10 * (wx1 * wy0)
                  + v01 * (wx0 * wy1) + v11 * (wx1 * wy1);
        sampled[((size_t)(b * 2 + c)) * HWSZ + p] = out;
    }
}

// ---------------------------------------------------------------------------
// Kernel 5: conv7x7 (2 -> 1, pad 3) + sigmoid. One thread per output pixel.
// ---------------------------------------------------------------------------
__global__ void k_attn7x7(const float* __restrict__ sampled,
                          const float* __restrict__ attn_w,
                          float* __restrict__ out) {
    __shared__ float sw[2 * KWIN * KWIN];
    for (int t = threadIdx.x; t < 2 * KWIN * KWIN; t += blockDim.x) sw[t] = attn_w[t];
    __syncthreads();

    int tid = blockIdx.x * blockDim.x + threadIdx.x;        // BATCH*HWSZ
    int b  = tid / HWSZ;
    int p  = tid - b * HWSZ;
    int hh = p / IMW;
    int ww = p - hh * IMW;

    float acc = 0.0f;
    #pragma unroll
    for (int c = 0; c < 2; ++c) {
        const float* fp = sampled + ((size_t)(b * 2 + c)) * HWSZ;
        for (int ky = 0; ky < KWIN; ++ky) {
            int y = hh + ky - 3;
            if ((unsigned)y >= (unsigned)IMH) continue;
            #pragma unroll
            for (int kx = 0; kx < KWIN; ++kx) {
                int x = ww + kx - 3;
                float v = ((unsigned)x < (unsigned)IMW) ? fp[(size_t)y * IMW + x] : 0.0f;
                acc = fmaf(v, sw[c * 49 + ky * KWIN + kx], acc);
            }
        }
    }
    out[tid] = 1.0f / (1.0f + expf(-acc));
}

// ---------------------------------------------------------------------------
// Launch
// ---------------------------------------------------------------------------
extern "C" void kernel_launch(void* const* d_in, const int* in_sizes, int n_in,
                              void* d_out, int out_size, void* d_ws, size_t ws_size,
                              hipStream_t stream) {
    (void)in_sizes; (void)n_in; (void)out_size; (void)ws_size;

    const float* x     = (const float*)d_in[0];
    const float* w1    = (const float*)d_in[1];
    const float* bn_g  = (const float*)d_in[2];
    const float* bn_b  = (const float*)d_in[3];
    const float* bn_m  = (const float*)d_in[4];
    const float* bn_v  = (const float*)d_in[5];
    const float* w2    = (const float*)d_in[6];
    const float* b2    = (const float*)d_in[7];
    const float* aw    = (const float*)d_in[8];
    float* out = (float*)d_out;

    // Workspace layout (bytes)
    char* ws = (char*)d_ws;
    const size_t sfBytes  = (size_t)BATCH * 2 * HWSZ * sizeof(float);   //  2 MB
    const size_t h1Bytes  = (size_t)BATCH * 16 * HWSZ * sizeof(float);  // 16 MB
    const size_t smBytes  = (size_t)BATCH * 2 * HWSZ * sizeof(float);   //  2 MB
    float* sf      = (float*)(ws);
    float* h1      = (float*)(ws + sfBytes);
    float* sampled = (float*)(ws + sfBytes + h1Bytes);
    float* w_eff   = (float*)(ws + sfBytes + h1Bytes + smBytes);
    float* b_eff   = w_eff + 288;

    // 1) channel mean/max reduce: 268 MB streaming read
    {
        int total = BATCH * (HWSZ / 4);                     // 65536
        k_chan_reduce<<<total / 256, 256, 0, stream>>>(x, sf);
    }
    // 2) fold mean-over-49 into conv2 weights
    k_fold_w2<<<1, 320, 0, stream>>>(w2, b2, w_eff, b_eff);
    // 3) conv1 + BN + ReLU via fp32 WMMA (16384 waves)
    {
        int waves  = BATCH * IMH * (IMW / 16);              // 16384
        int blocks = (waves * 32) / 256;                    // 2048
        k_conv1_wmma<<<blocks, 256, 0, stream>>>(sf, w1, bn_g, bn_b, bn_m, bn_v, h1);
    }
    // 4) folded offset conv + tanh + deformable bilinear sample
    {
        int total = BATCH * HWSZ;                           // 262144
        k_offset_sample<<<total / 256, 256, 0, stream>>>(h1, w_eff, b_eff, sf, sampled);
    }
    // 5) 7x7 attention conv + sigmoid
    {
        int total = BATCH * HWSZ;
        k_attn7x7<<<total / 256, 256, 0, stream>>>(sampled, aw, out);
    }
}